// STConv_56040733278959
// MI455X (gfx1250) — compile-verified
//
#include <hip/hip_runtime.h>
#include <hip/hip_bf16.h>

// ---------------- problem constants ----------------
#define NB   8
#define NT   16
#define NN   2048
#define CIN  2
#define HH   64
#define CO   64
#define KT   3
#define NE   32768
#define T1   14            // NT - KT + 1
#define T2   12            // T1 - KT + 1
#define NG   (NB * T1)     // 112 independent graphs for cheb_conv

typedef __attribute__((ext_vector_type(2))) float v2f;
typedef __attribute__((ext_vector_type(8))) float v8f;

// ---------------- elementwise helpers ----------------
__global__ void k_zero(float* __restrict__ p, int n) {
    int i = blockIdx.x * blockDim.x + threadIdx.x;
    if (i < n) p[i] = 0.f;
}

__global__ void k_negcopy(const float* __restrict__ src, float* __restrict__ dst, int n) {
    int i = blockIdx.x * blockDim.x + threadIdx.x;
    if (i < n) dst[i] = -src[i];
}

// ---------------- temporal conv 1 + GLU (C_IN=2 -> 2*H, tiny K) ----------------
__global__ void k_tconv1(const float* __restrict__ X, const float* __restrict__ W,
                         const float* __restrict__ bias, float* __restrict__ t0) {
    int idx = blockIdx.x * blockDim.x + threadIdx.x;   // NB*T1*NN*HH threads
    if (idx >= NB * T1 * NN * HH) return;
    int c = idx & (HH - 1);
    int n = (idx >> 6) & (NN - 1);
    int t = (idx / (HH * NN)) % T1;
    int b = idx / (HH * NN * T1);

    float yp = bias[c], yq = bias[c + HH];
#pragma unroll
    for (int kt = 0; kt < KT; ++kt) {
        const float* xp = X + (((long long)b * NT + t + kt) * NN + n) * CIN;
        float x0 = xp[0], x1 = xp[1];
        // conv1_w layout (2H, CIN, KT, 1): w[co][ci][kt]
        yp += x0 * W[(c * CIN + 0) * KT + kt]        + x1 * W[(c * CIN + 1) * KT + kt];
        yq += x0 * W[((c + HH) * CIN + 0) * KT + kt] + x1 * W[((c + HH) * CIN + 1) * KT + kt];
    }
    float xin = (c < CIN) ? X[(((long long)b * NT + t + KT - 1) * NN + n) * CIN + c] : 0.f;
    t0[idx] = (yp + xin) * (1.f / (1.f + __expf(-yq)));
}

// ---------------- graph normalization ----------------
__global__ void k_deg(const long long* __restrict__ ei, const float* __restrict__ ew,
                      float* __restrict__ deg) {
    int e = blockIdx.x * blockDim.x + threadIdx.x;
    if (e >= NE) return;
    int s = (int)ei[e];
    int d = (int)ei[NE + e];
    float w = (s == d) ? 0.f : ew[e];
    atomicAdd(&deg[s], w);
}

__global__ void k_wn(const long long* __restrict__ ei, const float* __restrict__ ew,
                     const float* __restrict__ deg, float* __restrict__ wn) {
    int e = blockIdx.x * blockDim.x + threadIdx.x;
    if (e >= NE) return;
    int s = (int)ei[e];
    int d = (int)ei[NE + e];
    float w  = (s == d) ? 0.f : ew[e];
    float ds = (deg[s] > 0.f) ? rsqrtf(deg[s]) : 0.f;
    float dd = (deg[d] > 0.f) ? rsqrtf(deg[d]) : 0.f;
    wn[e] = -ds * w * dd;
}

// ---------------- lhat scatter: out[g,dst,f] += scale * in[g,src,f] * wn[e] ----------------
__global__ void k_scatter(const float* __restrict__ in, float* __restrict__ out,
                          const long long* __restrict__ ei, const float* __restrict__ wn,
                          float scale) {
    int idx = blockIdx.x * blockDim.x + threadIdx.x;  // NG*NE*HH threads
    int f = idx & (HH - 1);
    int e = (idx >> 6) & (NE - 1);
    int g = idx >> 21;                                // / (HH*NE)
    if (g >= NG) return;
    float w = wn[e] * scale;
    if (w == 0.f) return;                             // uniform per wave (f spans one edge)
    int s = (int)ei[e];
    int d = (int)ei[NE + e];
    float v = in[((long long)g * NN + s) * HH + f] * w;
    atomicAdd(&out[((long long)g * NN + d) * HH + f], v);
}

// ---------------- Cheb GEMM: tg = relu(tx0@W0 + tx1@W1 + tx2@W2 + b) ----------------
// one wave per (graph, 16-row M tile); 4 accumulators cover all 64 output features.
// Weights staged once per block into LDS in WMMA-fragment order (paired K values).
#define CHEB_FRAGS (3 * 16 * 4 * 32)   // mat x kkstep x jtile x lane = 6144 float2 = 48 KB
__global__ void k_cheb_gemm(const float* __restrict__ tx0, const float* __restrict__ tx1,
                            const float* __restrict__ tx2, const float* __restrict__ W,
                            const float* __restrict__ bias, float* __restrict__ tg) {
    __shared__ v2f ldsB[CHEB_FRAGS];

    // ---- cooperative stage: repack cheb_w[mat][h_in][h_out] into per-lane fragments ----
    for (int i = threadIdx.x; i < CHEB_FRAGS; i += 256) {
        int ln   = i & 31;
        int j    = (i >> 5) & 3;
        int kk4  = (i >> 7) & 15;
        int mat  = i >> 11;
        int khL  = (ln >> 4) << 1;
        int col  = j * 16 + (ln & 15);
        int row  = kk4 * 4 + khL;
        const float* Bm = W + mat * HH * HH;
        v2f frag;
        frag.x = Bm[row * HH + col];
        frag.y = Bm[(row + 1) * HH + col];
        ldsB[i] = frag;
    }
    __syncthreads();

    int wave  = (blockIdx.x * blockDim.x + threadIdx.x) >> 5;   // exact grid: NG*128 waves
    int lane  = threadIdx.x & 31;
    int mtile = wave & 127;                                     // NN/16 = 128
    int g     = wave >> 7;
    int m     = lane & 15;
    int kh    = (lane >> 4) << 1;                               // K-half offset: 0 or 2

    const float* txs[3] = {tx0, tx1, tx2};
    v8f acc[4] = {};
#pragma unroll
    for (int mat = 0; mat < 3; ++mat) {
        const float* A = txs[mat] + ((long long)g * NN + mtile * 16 + m) * HH;
        for (int kk4 = 0; kk4 < 16; ++kk4) {
            v2f a = *(const v2f*)(A + kk4 * 4 + kh);
            const v2f* brow = &ldsB[((mat * 16 + kk4) * 4) * 32 + lane];
#pragma unroll
            for (int j = 0; j < 4; ++j) {
                v2f b = brow[j * 32];                            // ds_load_b64
                acc[j] = __builtin_amdgcn_wmma_f32_16x16x4_f32(
                    false, a, false, b, (short)0, acc[j], false, false);
            }
        }
    }
#pragma unroll
    for (int j = 0; j < 4; ++j)
#pragma unroll
        for (int v = 0; v < 8; ++v) {
            int row = mtile * 16 + ((lane < 16) ? v : v + 8);
            int col = j * 16 + m;
            float val = acc[j][v] + bias[col];
            tg[((long long)g * NN + row) * HH + col] = fmaxf(val, 0.f);
        }
}

// ---------------- transpose conv2 weights -> Wt[kt][h][co] ----------------
__global__ void k_wt(const float* __restrict__ w2, float* __restrict__ wt) {
    int i = blockIdx.x * blockDim.x + threadIdx.x;   // KT*HH*128
    if (i >= KT * HH * 128) return;
    int co = i & 127;
    int h  = (i >> 7) & (HH - 1);
    int kt = i / (128 * HH);
    wt[i] = w2[(co * HH + h) * KT + kt];             // conv2_w (2*CO, HH, KT, 1)
}

// ---------------- temporal conv 2 (K=192 GEMM) + GLU ----------------
// one wave per (b, t_out, 16-row M tile); 8 accumulators cover 128 channels (p|q).
// Transposed weights staged per block into LDS in WMMA-fragment order (96 KB).
#define TC2_FRAGS (3 * 16 * 8 * 32)    // kt x kkstep x jtile x lane = 12288 float2 = 96 KB
__global__ void k_tconv2(const float* __restrict__ tg, const float* __restrict__ Wt,
                         const float* __restrict__ bias, float* __restrict__ out) {
    __shared__ v2f ldsB[TC2_FRAGS];

    // ---- cooperative stage: repack Wt[kt][h][co] into per-lane fragments ----
    for (int i = threadIdx.x; i < TC2_FRAGS; i += 256) {
        int ln   = i & 31;
        int j    = (i >> 5) & 7;
        int kk4  = (i >> 8) & 15;
        int kt   = i >> 12;
        int khL  = (ln >> 4) << 1;
        int col  = j * 16 + (ln & 15);
        int row  = kk4 * 4 + khL;
        const float* Bm = Wt + kt * HH * 128;
        v2f frag;
        frag.x = Bm[row * 128 + col];
        frag.y = Bm[(row + 1) * 128 + col];
        ldsB[i] = frag;
    }
    __syncthreads();

    int wave  = (blockIdx.x * blockDim.x + threadIdx.x) >> 5;   // exact grid: NB*T2*128 waves
    int lane  = threadIdx.x & 31;
    int mtile = wave & 127;
    int t     = (wave >> 7) % T2;
    int b     = wave / (128 * T2);
    int m     = lane & 15;
    int kh    = (lane >> 4) << 1;

    v8f acc[8] = {};
#pragma unroll
    for (int kt = 0; kt < KT; ++kt) {
        const float* A = tg + (((long long)b * T1 + t + kt) * NN + mtile * 16 + m) * HH;
        for (int kk4 = 0; kk4 < 16; ++kk4) {
            v2f a = *(const v2f*)(A + kk4 * 4 + kh);
            const v2f* brow = &ldsB[((kt * 16 + kk4) * 8) * 32 + lane];
#pragma unroll
            for (int j = 0; j < 8; ++j) {
                v2f bb = brow[j * 32];                           // ds_load_b64
                acc[j] = __builtin_amdgcn_wmma_f32_16x16x4_f32(
                    false, a, false, bb, (short)0, acc[j], false, false);
            }
        }
    }
    // GLU: p = acc[0..3] (+bias+residual), q = acc[4..7]
#pragma unroll
    for (int j = 0; j < 4; ++j)
#pragma unroll
        for (int v = 0; v < 8; ++v) {
            int row = mtile * 16 + ((lane < 16) ? v : v + 8);
            int col = j * 16 + m;
            float xin = tg[(((long long)b * T1 + t + 2) * NN + row) * HH + col];
            float p = acc[j][v]     + bias[col]      + xin;
            float q = acc[j + 4][v] + bias[col + CO];
            out[(((long long)b * T2 + t) * NN + row) * CO + col] =
                p * (1.f / (1.f + __expf(-q)));
        }
}

// ---------------- host launcher ----------------
extern "C" void kernel_launch(void* const* d_in, const int* in_sizes, int n_in,
                              void* d_out, int out_size, void* d_ws, size_t ws_size,
                              hipStream_t stream) {
    (void)in_sizes; (void)n_in; (void)out_size; (void)ws_size;
    const float*     X    = (const float*)d_in[0];
    const long long* ei   = (const long long*)d_in[1];   // int64 edge_index (2, E)
    const float*     ew   = (const float*)d_in[2];
    const float*     w1   = (const float*)d_in[3];
    const float*     b1   = (const float*)d_in[4];
    const float*     wc   = (const float*)d_in[5];
    const float*     bc   = (const float*)d_in[6];
    const float*     w2   = (const float*)d_in[7];
    const float*     b2   = (const float*)d_in[8];
    float*           outp = (float*)d_out;

    const long long NFEAT = (long long)NG * NN * HH;     // 14,680,064 floats per tensor
    float* t0  = (float*)d_ws;
    float* tx1 = t0  + NFEAT;
    float* tx2 = tx1 + NFEAT;
    float* tg  = tx2 + NFEAT;
    float* wn  = tg  + NFEAT;
    float* deg = wn  + NE;
    float* w2t = deg + NN;

    const int TB = 256;
    int nfeat_blocks = (int)((NFEAT + TB - 1) / TB);     // 57344
    long long scat_total = (long long)NG * NE * HH;
    int scat_blocks = (int)((scat_total + TB - 1) / TB); // 917504

    // 1) temporal conv 1 + GLU
    k_tconv1<<<nfeat_blocks, TB, 0, stream>>>(X, w1, b1, t0);

    // 2) edge normalization
    k_zero<<<(NN + TB - 1) / TB, TB, 0, stream>>>(deg, NN);
    k_deg<<<(NE + TB - 1) / TB, TB, 0, stream>>>(ei, ew, deg);
    k_wn<<<(NE + TB - 1) / TB, TB, 0, stream>>>(ei, ew, deg, wn);

    // 3) Chebyshev recursion via scatters
    k_zero<<<nfeat_blocks, TB, 0, stream>>>(tx1, (int)NFEAT);
    k_scatter<<<scat_blocks, TB, 0, stream>>>(t0, tx1, ei, wn, 1.0f);
    k_negcopy<<<nfeat_blocks, TB, 0, stream>>>(t0, tx2, (int)NFEAT);
    k_scatter<<<scat_blocks, TB, 0, stream>>>(tx1, tx2, ei, wn, 2.0f);

    // 4) Cheb GEMM (WMMA fp32, LDS-staged weights) + bias + ReLU
    {
        int waves = NG * (NN / 16);                      // 14336
        k_cheb_gemm<<<waves * 32 / TB, TB, 0, stream>>>(t0, tx1, tx2, wc, bc, tg);
    }

    // 5) temporal conv 2 (WMMA fp32, K=192, LDS-staged weights) + GLU
    k_wt<<<(KT * HH * 128 + TB - 1) / TB, TB, 0, stream>>>(w2, w2t);
    {
        int waves = NB * T2 * (NN / 16);                 // 12288
        k_tconv2<<<waves * 32 / TB, TB, 0, stream>>>(tg, w2t, b2, outp);
    }
}